// BayesianTorchModel_37022618092109
// MI455X (gfx1250) — compile-verified
//
#include <hip/hip_runtime.h>

typedef __attribute__((ext_vector_type(2))) float v2f;
typedef __attribute__((ext_vector_type(8))) float v8f;

static constexpr int kM    = 8192;  // nodes per layer
static constexpr int kK    = 8;     // parents per node
static constexpr int kC    = 256;   // CPT configs per node
static constexpr int kLay  = 8;     // internal layers
static constexpr int kWavesPerBlock = 4;
static constexpr int kNodesPerBlock = kWavesPerBlock * 16;

__device__ __forceinline__ float fast_sigmoid(float x) {
    // 1 / (1 + e^-x): native v_exp_f32 + v_rcp_f32
    return __builtin_amdgcn_rcpf(1.0f + __expf(-x));
}

__device__ __forceinline__ float pick8(v8f c, int i) {
    float r = c[0];
    if (i == 1) r = c[1];
    if (i == 2) r = c[2];
    if (i == 3) r = c[3];
    if (i == 4) r = c[4];
    if (i == 5) r = c[5];
    if (i == 6) r = c[6];
    if (i == 7) r = c[7];
    return r;
}

// Roots: out[i] = sigmoid(root_logits[i])
__global__ void bn_root_kernel(const float* __restrict__ logits,
                               float* __restrict__ out) {
    int i = blockIdx.x * blockDim.x + threadIdx.x;
    if (i < kM) out[i] = fast_sigmoid(logits[i]);
}

// One internal layer. Each wave handles 16 nodes.
// Lane l: node n = l&15, config-half h = l>>4 (configs == {2h, 2h+1} mod 4).
// For chunk t (configs 4t..4t+3):
//   A (16x4 f32):  lanes 0-15 hold K=0,1; lanes 16-31 hold K=2,3  -> {prod[c0], prod[c0+1]}
//   B (4x16 f32):  same lane structure with N = node                -> {sig[c0], sig[c0+1]}
// Diagonal of 16x16 f32 D: lanes 0-7 elem=lane (m=lane), lanes 24-31 elem=lane-24 (m=lane-16).
__global__ __launch_bounds__(kWavesPerBlock * 32)
void bn_layer_kernel(const float* __restrict__ cpt,    // [kM][kC]
                     const int*   __restrict__ par,    // [kM][kK]
                     const float* __restrict__ prev,   // [kM]
                     float*       __restrict__ outm) { // [kM]
    const int lane = threadIdx.x & 31;
    const int wave = threadIdx.x >> 5;
    const int n    = lane & 15;
    const int h    = lane >> 4;
    const int tile = blockIdx.x * kWavesPerBlock + wave;
    const int node = tile * 16 + n;

    // Gather this node's 8 parent marginals from the previous layer.
    const int* pp = par + node * kK;
    float p[kK], q[kK];
#pragma unroll
    for (int j = 0; j < kK; ++j) {
        p[j] = prev[pp[j]];
        q[j] = 1.0f - p[j];
    }
    // c0 = 4t + 2h: bit0(c0)=0 (parent 7 factor split out), bit1(c0)=h -> parent 6.
    const float fh = h ? p[6] : q[6];

    const float* row = cpt + (size_t)node * kC + 2 * h;
    __builtin_prefetch(row, 0, 3);            // global_prefetch_b8
    __builtin_prefetch(row + kC / 2, 0, 3);

    v8f acc = {};
#pragma unroll
    for (int hi = 0; hi < 8; ++hi) {          // t[5:3] -> parents 0,1,2
        const float rH = (((hi >> 2) & 1) ? p[0] : q[0])
                       * (((hi >> 1) & 1) ? p[1] : q[1])
                       * (((hi     ) & 1) ? p[2] : q[2]) * fh;
#pragma unroll
        for (int lo = 0; lo < 8; ++lo) {      // t[2:0] -> parents 3,4,5
            const int t = hi * 8 + lo;
            const float rL = (((lo >> 2) & 1) ? p[3] : q[3])
                           * (((lo >> 1) & 1) ? p[4] : q[4])
                           * (((lo     ) & 1) ? p[5] : q[5]);
            const float r7 = rH * rL;

            v2f a;
            a.x = r7 * q[7];                  // config c0   (parent-7 bit = 0)
            a.y = r7 * p[7];                  // config c0+1 (parent-7 bit = 1)

            const float2 cl = *(const float2*)(row + 4 * t);
            v2f b;
            b.x = fast_sigmoid(cl.x);
            b.y = fast_sigmoid(cl.y);

            // D = A*B + C, full fp32 matrix op; we consume only the diagonal.
            acc = __builtin_amdgcn_wmma_f32_16x16x4_f32(
                false, a, false, b, (short)0, acc, false, false);
        }
    }

    // Diagonal extraction + store (EXEC divergence only after all WMMAs).
    if (lane < 8) {
        outm[tile * 16 + lane] = pick8(acc, lane);
    } else if (lane >= 24) {
        outm[tile * 16 + (lane - 16)] = pick8(acc, lane - 24);
    }
}

extern "C" void kernel_launch(void* const* d_in, const int* in_sizes, int n_in,
                              void* d_out, int out_size, void* d_ws, size_t ws_size,
                              hipStream_t stream) {
    (void)in_sizes; (void)n_in; (void)out_size; (void)d_ws; (void)ws_size;
    const float* root    = (const float*)d_in[0];
    const float* cpt     = (const float*)d_in[1];
    const int*   parents = (const int*)d_in[2];
    float* out = (float*)d_out;

    bn_root_kernel<<<kM / 256, 256, 0, stream>>>(root, out);

    for (int l = 0; l < kLay; ++l) {
        bn_layer_kernel<<<kM / kNodesPerBlock, kWavesPerBlock * 32, 0, stream>>>(
            cpt + (size_t)l * kM * kC,
            parents + (size_t)l * kM * kK,
            out + (size_t)l * kM,
            out + (size_t)(l + 1) * kM);
    }
}